// MyGlobalAttention_89541478187114
// MI455X (gfx1250) — compile-verified
//
#include <hip/hip_runtime.h>
#include <hip/hip_bf16.h>
#include <math.h>

typedef float v2f __attribute__((ext_vector_type(2)));
typedef float v8f __attribute__((ext_vector_type(8)));

#define N_GRAPHS 512
#define F 256
#define CHUNK 512
// encoding of -inf under the order-preserving float->uint map
#define ENC_NEG_INF 0x007FFFFFu

// Monotonic float<->uint mapping so unsigned atomicMax == float max.
__device__ __forceinline__ unsigned enc_f32(float f) {
    unsigned u = __float_as_uint(f);
    return (u & 0x80000000u) ? ~u : (u | 0x80000000u);
}
__device__ __forceinline__ float dec_f32(unsigned e) {
    unsigned u = (e & 0x80000000u) ? (e ^ 0x80000000u) : ~e;
    return __uint_as_float(u);
}

// ---------------- Pass 0: init scratch (must rerun every launch) ------------
__global__ void ga_init_kernel(float* __restrict__ z,
                               float* __restrict__ denom,
                               unsigned* __restrict__ segmax) {
    int i = blockIdx.x * blockDim.x + threadIdx.x;
    if (i < N_GRAPHS * F) z[i] = 0.0f;
    if (i < N_GRAPHS) { denom[i] = 0.0f; segmax[i] = ENC_NEG_INF; }
}

// ---------------- Pass 1: gate_raw = x @ W_gate + b; segment max ------------
// One wave32 per node; lane j reads x[i][j], x[i][j+32], ... (128B coalesced).
__global__ void ga_gate_dot_kernel(const float* __restrict__ x,
                                   const int* __restrict__ batch,
                                   const float* __restrict__ Wg,
                                   const float* __restrict__ bg,
                                   float* __restrict__ gate_raw,
                                   unsigned* __restrict__ segmax,
                                   int n) {
    int wave = threadIdx.x >> 5;
    int lane = threadIdx.x & 31;
    int i = blockIdx.x * 8 + wave;
    if (i >= n) return;
    const float* row = x + (size_t)i * F;
    float acc = 0.0f;
#pragma unroll
    for (int j = 0; j < F; j += 32)
        acc = fmaf(row[j + lane], Wg[j + lane], acc);
#pragma unroll
    for (int off = 16; off >= 1; off >>= 1)
        acc += __shfl_xor(acc, off, 32);
    if (lane == 0) {
        float g = acc + bg[0];
        gate_raw[i] = g;
        atomicMax(&segmax[batch[i]], enc_f32(g));
    }
}

// ---------------- Pass 2: e = exp(g - segmax); denom += e -------------------
__global__ void ga_exp_sum_kernel(const int* __restrict__ batch,
                                  const unsigned* __restrict__ segmax,
                                  float* __restrict__ gate_e,  // in: raw, out: e
                                  float* __restrict__ denom,
                                  int n) {
    int i = blockIdx.x * blockDim.x + threadIdx.x;
    if (i >= n) return;
    int b = batch[i];
    float e = expf(gate_e[i] - dec_f32(segmax[b]));
    gate_e[i] = e;
    atomicAdd(&denom[b], e);
}

// ---------------- Pass 3: gate = e / (denom + 1e-16) -> output slot ---------
__global__ void ga_gate_norm_kernel(const int* __restrict__ batch,
                                    const float* __restrict__ denom,
                                    const float* __restrict__ gate_e,
                                    float* __restrict__ gate_out,
                                    int n) {
    int i = blockIdx.x * blockDim.x + threadIdx.x;
    if (i >= n) return;
    gate_out[i] = gate_e[i] / (denom[batch[i]] + 1e-16f);
}

// ---------------- Pass 4: z[s][f] += gate_i * x[i][f] -----------------------
// 256 threads = one feature each; batch is sorted so a register accumulator
// is flushed to global z only on segment change (~2 atomics / chunk / thread).
__global__ void ga_weighted_segsum_kernel(const float* __restrict__ x,
                                          const int* __restrict__ batch,
                                          const float* __restrict__ gate,
                                          float* __restrict__ z,
                                          int n) {
    int f = threadIdx.x;  // 0..255
    int start = blockIdx.x * CHUNK;
    if (start >= n) return;
    int end = min(start + CHUNK, n);
    int cur = batch[start];
    float acc = 0.0f;
    for (int i = start; i < end; ++i) {
        if (i + 8 < end)
            __builtin_prefetch(&x[(size_t)(i + 8) * F + f], 0, 0);
        int b = batch[i];
        if (b != cur) {
            atomicAdd(&z[(size_t)cur * F + f], acc);
            acc = 0.0f;
            cur = b;
        }
        acc = fmaf(gate[i], x[(size_t)i * F + f], acc);
    }
    atomicAdd(&z[(size_t)cur * F + f], acc);
}

// ---------------- Pass 5: out = z @ W_nn + b_nn via fp32 WMMA ---------------
// One wave per 16x16 output tile; K swept 4 at a time with
// v_wmma_f32_16x16x4_f32 (full fp32 precision on the matrix pipes).
// A layout (16x4): lane = {half=l>>4, lr=l&15}; holds A[lr][2*half+{0,1}].
// B layout (4x16): lane holds B[2*half+{0,1}][lr].
// D layout: VGPR v, lanes 0-15 -> M=v, lanes 16-31 -> M=8+v; N = lr.
__global__ void ga_gemm_wmma_kernel(const float* __restrict__ z,
                                    const float* __restrict__ Wnn,
                                    const float* __restrict__ bnn,
                                    float* __restrict__ out) {
    int waveId = blockIdx.x * (blockDim.x >> 5) + (threadIdx.x >> 5);
    const int NT = F / 16;  // 16 tiles along F_OUT
    int mt = waveId / NT;   // 0..31
    int nt = waveId % NT;   // 0..15
    int lane = threadIdx.x & 31;
    int half = lane >> 4;
    int lr = lane & 15;
    int m = mt * 16 + lr;
    int n = nt * 16 + lr;

    v8f c = {};
#pragma unroll 4
    for (int k = 0; k < F; k += 4) {
        int kb = k + half * 2;
        v2f a, b;
        a.x = z[(size_t)m * F + kb];
        a.y = z[(size_t)m * F + kb + 1];
        b.x = Wnn[(size_t)kb * F + n];
        b.y = Wnn[(size_t)(kb + 1) * F + n];
        c = __builtin_amdgcn_wmma_f32_16x16x4_f32(
                /*neg_a=*/false, a, /*neg_b=*/false, b,
                /*c_mod=*/(short)0, c, /*reuse_a=*/false, /*reuse_b=*/false);
    }
    float bias = bnn[n];
#pragma unroll
    for (int v = 0; v < 8; ++v) {
        int row = mt * 16 + half * 8 + v;
        out[(size_t)row * F + nt * 16 + lr] = c[v] + bias;
    }
}

extern "C" void kernel_launch(void* const* d_in, const int* in_sizes, int n_in,
                              void* d_out, int out_size, void* d_ws, size_t ws_size,
                              hipStream_t stream) {
    const float* x      = (const float*)d_in[0];
    const int*   batch  = (const int*)d_in[1];
    // d_in[2] = size scalar (compile-time N_GRAPHS=512)
    const float* W_gate = (const float*)d_in[3];
    const float* b_gate = (const float*)d_in[4];
    const float* W_nn   = (const float*)d_in[5];
    const float* b_nn   = (const float*)d_in[6];

    const int n = in_sizes[1];  // number of nodes

    float* out_mat  = (float*)d_out;                       // [512, 256]
    float* gate_out = (float*)d_out + (size_t)N_GRAPHS * F;  // [N, 1]

    // Workspace layout
    char* ws = (char*)d_ws;
    float*    gate_e = (float*)ws;                                // [N]
    unsigned* segmax = (unsigned*)(ws + (size_t)n * 4);           // [512]
    float*    denom  = (float*)(ws + (size_t)n * 4 + 2048);       // [512]
    float*    z      = (float*)(ws + (size_t)n * 4 + 4096);       // [512,256]

    // Pass 0: init scratch
    ga_init_kernel<<<(N_GRAPHS * F + 255) / 256, 256, 0, stream>>>(z, denom, segmax);
    // Pass 1: gate dot + segment max (8 waves/block, one node per wave)
    ga_gate_dot_kernel<<<(n + 7) / 8, 256, 0, stream>>>(
        x, batch, W_gate, b_gate, gate_e, segmax, n);
    // Pass 2: exp + denom
    ga_exp_sum_kernel<<<(n + 255) / 256, 256, 0, stream>>>(
        batch, segmax, gate_e, denom, n);
    // Pass 3: normalize gate into output
    ga_gate_norm_kernel<<<(n + 255) / 256, 256, 0, stream>>>(
        batch, denom, gate_e, gate_out, n);
    // Pass 4: z[s] = sum gate_i * x_i  (second streaming pass over x)
    ga_weighted_segsum_kernel<<<(n + CHUNK - 1) / CHUNK, 256, 0, stream>>>(
        x, batch, gate_out, z, n);
    // Pass 5: out = z @ W_nn + b_nn  (512 waves = 512 tiles, 8 waves/block)
    ga_gemm_wmma_kernel<<<(N_GRAPHS / 16) * (F / 16) / 8, 256, 0, stream>>>(
        z, W_nn, b_nn, out_mat);
}